// NeedlemanWunschDecoder_72155450573572
// MI455X (gfx1250) — compile-verified
//
#include <hip/hip_runtime.h>

// Smoothed Needleman-Wunsch wavefront DP for 2048x2048 on gfx1250.
// One wave32 per 32-row band; skewed lane-per-row wavefront with register
// shuffles. theta/A tiles staged into LDS by the Tensor Data Mover
// (tensor_load_to_lds, TENSORcnt), triple-buffered and overlapped with
// compute. Cross-band dependency pipelined via release/acquire atomics.

#define NN      2048
#define BANDS   64
#define CHUNK   64
#define NCHUNK  32            // 2048 / 64
#define RSTRIDE 65            // 64 data dwords + 1 pad dword (TDM pad_enable)
#define SLOT_F  (32 * RSTRIDE)      // 2080 floats per tile slot
#define THETA_OFF 0
#define A_OFF     (3 * SLOT_F)      // 6240
#define ROW_OFF   (6 * SLOT_F)      // 12480
#define SMEM_FLOATS (ROW_OFF + 128) // 12608
#define SMEM_BYTES  (SMEM_FLOATS * 4) // 50432 B dynamic LDS

typedef unsigned int u32x4 __attribute__((ext_vector_type(4)));
typedef int          i32x8 __attribute__((ext_vector_type(8)));
typedef int          i32x4 __attribute__((ext_vector_type(4)));

// Issue one TDM load of a 32-row x 64-col float tile (row stride 2048 floats)
// into LDS at lds_byte_off, with 1-dword padding after every 64 dwords so the
// LDS row stride is 65 floats (bank-conflict-free column reads).
__device__ __forceinline__ void tdm_tile_load(const float* gptr, unsigned lds_byte_off) {
  unsigned long long ga = (unsigned long long)gptr;
  u32x4 g0;
  g0[0] = 1u;                                   // count=1, user mode, no gather
  g0[1] = lds_byte_off;                         // lds_addr
  g0[2] = (unsigned)ga;                         // global_addr[31:0]
  g0[3] = ((unsigned)(ga >> 32) & 0x01FFFFFFu)  // global_addr[56:32]
        | 0x80000000u;                          // type=2 ("image")
  i32x8 g1;
  g1[0] = 0x01520000;          // data_size=4B, pad_enable, pad_interval=64dw, pad_amount=1dw
  g1[1] = (int)(2048u << 16);  // tensor_dim0[15:0] in [31:16]
  g1[2] = (int)(2048u << 16);  // tensor_dim0 hi=0 | tensor_dim1[15:0]
  g1[3] = (int)(64u << 16);    // tensor_dim1 hi=0 | tile_dim0=64
  g1[4] = 32;                  // tile_dim1=32, tile_dim2=0
  g1[5] = 2048;                // tensor_dim0_stride low (floats)
  g1[6] = 0;
  g1[7] = 0;
  i32x4 gz4 = {0, 0, 0, 0};            // groups 2/3 unused (2D tensor)
  i32x8 gz8 = {0, 0, 0, 0, 0, 0, 0, 0};
  __builtin_amdgcn_tensor_load_to_lds(g0, g1, gz4, gz4, gz8, 0);
}

__global__ void nw_init(int* __restrict__ progress) {
  if (threadIdx.x < BANDS) progress[threadIdx.x] = 0;
}

__global__ void __launch_bounds__(32)
nw_wavefront(const float* __restrict__ theta, const float* __restrict__ Amat,
             float* __restrict__ rowBuf, int* __restrict__ progress,
             float* __restrict__ out) {
  extern __shared__ float smem[];
  const int lane = threadIdx.x & 31;
  const int band = blockIdx.x;
  const int row0 = band * 32;

  // Zero both boundary-row LDS buffers (band 0 reads these as V[0,:] = 0).
  smem[ROW_OFF + lane]      = 0.f;
  smem[ROW_OFF + 32 + lane] = 0.f;
  smem[ROW_OFF + 64 + lane] = 0.f;
  smem[ROW_OFF + 96 + lane] = 0.f;

  // Prologue: DMA chunk 0 of theta/A into slot 0.
  tdm_tile_load(theta + (size_t)row0 * NN, (unsigned)((THETA_OFF) * 4));
  tdm_tile_load(Amat  + (size_t)row0 * NN, (unsigned)((A_OFF) * 4));
  __builtin_amdgcn_s_wait_tensorcnt(0);

  const float* prevRow  = rowBuf + (size_t)(band - 1) * NN;
  float*       myRow    = rowBuf + (size_t)band * NN;
  int*         prevProg = progress + (band - 1);

  float v = 0.f, h1 = 0.f, h2 = 0.f, rp_prev = 0.f;

  for (int g = 0; g <= NCHUNK; ++g) {
    // Prefetch data chunk g+1 into slot (g+1)%3 (holds chunk g-2: free).
    if (g <= NCHUNK - 2) {
      const int s = (g + 1) % 3;
      tdm_tile_load(theta + (size_t)row0 * NN + (size_t)(g + 1) * CHUNK,
                    (unsigned)((THETA_OFF + s * SLOT_F) * 4));
      tdm_tile_load(Amat + (size_t)row0 * NN + (size_t)(g + 1) * CHUNK,
                    (unsigned)((A_OFF + s * SLOT_F) * 4));
    }
    // Acquire previous band's boundary-row chunk g and stage it in LDS.
    if (band != 0 && g <= NCHUNK - 1) {
      const int need = (g + 1) * CHUNK;
      while (__hip_atomic_load(prevProg, __ATOMIC_ACQUIRE,
                               __HIP_MEMORY_SCOPE_AGENT) < need)
        __builtin_amdgcn_s_sleep(2);
      const int base = g * CHUNK;
      const float r0 = prevRow[base + lane];
      const float r1 = prevRow[base + 32 + lane];
      const int dst = ROW_OFF + ((g & 1) << 6);
      smem[dst + lane]      = r0;
      smem[dst + 32 + lane] = r1;
    }

    const int slotCur  = g % 3;
    const int slotPrev = (g + 2) % 3;
    const int thCur  = THETA_OFF + slotCur  * SLOT_F + lane * RSTRIDE;
    const int thPrev = THETA_OFF + slotPrev * SLOT_F + lane * RSTRIDE;
    const int aCur   = A_OFF + slotCur  * SLOT_F + lane * RSTRIDE;
    const int aPrev  = A_OFF + slotPrev * SLOT_F + lane * RSTRIDE;
    const int rowLds = ROW_OFF + ((g & 1) << 6);
    const int gbase  = g * CHUNK;
    const int tEnd   = (g == NCHUNK) ? (NN + 31) : (gbase + CHUNK);

    for (int t = gbase; t < tEnd; ++t) {
      const int col = t - lane;                      // 0-indexed input column
      const bool active = (col >= 0) && (col < NN);
      const int off = col & (CHUNK - 1);
      const bool cur = (col >= gbase);
      const float th = smem[(cur ? thCur : thPrev) + off];
      const float a  = smem[(cur ? aCur  : aPrev)  + off];

      float up = __shfl_up(h1, 1, 32);   // V[i-1, j]   (lane-1, step t-1)
      float dg = __shfl_up(h2, 1, 32);   // V[i-1, j-1] (lane-1, step t-2)
      if (lane == 0) {                   // boundary row from band above
        const float rp = smem[rowLds + (t & (CHUNK - 1))];
        up = rp;
        dg = rp_prev;
        rp_prev = rp;
      }

      const float x  = a + up;           // insert
      const float zc = a + v;            // delete (v = V[i, j-1])
      float mx = fmaxf(fmaxf(x, dg), zc);
      float e = __builtin_amdgcn_exp2f((x  - mx) * 1.44269504f)
              + __builtin_amdgcn_exp2f((dg - mx) * 1.44269504f)
              + __builtin_amdgcn_exp2f((zc - mx) * 1.44269504f);
      const float vn = th + mx + __builtin_amdgcn_logf(e) * 0.69314718f;

      if (active) {
        if (lane == 31) {
          myRow[col] = vn;                              // bottom-row handoff
          if (band == BANDS - 1 && col == NN - 1) out[0] = vn;  // V[N, M]
        }
        h2 = h1; h1 = vn; v = vn;
      }
      // Publish progress (entries 0..t-31 written) every 32 steps; release
      // makes the myRow stores visible to the consumer's acquire.
      if (lane == 31 && (t & 31) == 30 && t >= 62) {
        __hip_atomic_store(progress + band, t - 30, __ATOMIC_RELEASE,
                           __HIP_MEMORY_SCOPE_AGENT);
      }
    }
    if (g <= NCHUNK - 2) __builtin_amdgcn_s_wait_tensorcnt(0);
  }
}

extern "C" void kernel_launch(void* const* d_in, const int* in_sizes, int n_in,
                              void* d_out, int out_size, void* d_ws, size_t ws_size,
                              hipStream_t stream) {
  (void)in_sizes; (void)n_in; (void)out_size; (void)ws_size;
  const float* theta = (const float*)d_in[0];
  const float* Amat  = (const float*)d_in[1];
  float* out = (float*)d_out;

  float* rowBuf  = (float*)d_ws;                                  // 64*2048 floats
  int*   progress = (int*)((char*)d_ws + (size_t)BANDS * NN * 4); // 64 ints

  nw_init<<<1, 64, 0, stream>>>(progress);
  nw_wavefront<<<BANDS, 32, SMEM_BYTES, stream>>>(theta, Amat, rowBuf, progress, out);
}